// GCN_87797721465132
// MI455X (gfx1250) — compile-verified
//
#include <hip/hip_runtime.h>

// ---------------------------------------------------------------------------
// GCN (10-layer) for MI455X / gfx1250.
//  - GEMM h@W on the matrix pipe: V_WMMA_F32_16X16X4_F32 (exact f32).
//    W is pre-packed into B-fragment order so every A/B load is a b64 with
//    compile-time strides; each wave computes a 16x64 strip (4 accumulators)
//    reusing its A fragment 4x.  Per k-step: 5 x b64 loads -> 4 WMMA.
//  - Edge aggregation: float4 gather (hW is 51MB, resident in 192MB L2) and
//    global_atomic_add_f32 scatter.
//  - Self-loop term folded into the bias/ReLU epilogue (no atomics).
// ---------------------------------------------------------------------------

typedef __attribute__((ext_vector_type(2))) float v2f;
typedef __attribute__((ext_vector_type(8))) float v8f;

// ---------------------------------------------------------------------------
// Pack W[128 x nCols] (row-major) into WMMA B-fragment order:
//   Wp[((tile*32 + ks)*32 + lane)*2 + {0,1}] = W[(4*ks + 2*(lane>>4) + {0,1})][tile*16 + (lane&15)]
// One thread per (tile, ks, lane) fragment pair.  K is fixed at 128 (32 ksteps).
// ---------------------------------------------------------------------------
__global__ void pack_W_frag(const float* __restrict__ W, float* __restrict__ Wp,
                            int nCols) {
  int idx = blockIdx.x * blockDim.x + threadIdx.x;
  int total = (nCols >> 4) << 10;            // tiles * 32 ksteps * 32 lanes
  if (idx >= total) return;
  int lane = idx & 31;
  int ks   = (idx >> 5) & 31;
  int tile = idx >> 10;
  int m = lane & 15;
  int half = lane >> 4;
  int ka = (ks << 2) + (half << 1);
  int col = (tile << 4) + m;
  Wp[2 * idx + 0] = W[ka * nCols + col];
  Wp[2 * idx + 1] = W[(ka + 1) * nCols + col];
}

// ---------------------------------------------------------------------------
// WMMA GEMM: Out[rows x NCOLS] = H[rows x 128] @ W[128 x NCOLS]   (f32)
// TPW = 16-col tiles per wave (A-fragment reuse factor).
// A layout (16x4 f32, ISA 7.12.2): lanes 0-15 -> M, lane half selects K pair.
// C/D layout: acc[v] = D[v + 8*(lane>>4)][lane&15].
// ---------------------------------------------------------------------------
template <int TPW, int NCOLS>
__global__ void gcn_gemm_wmma_f32(const float* __restrict__ H,
                                  const float* __restrict__ Wp,
                                  float* __restrict__ Out,
                                  int nRowTiles) {
  constexpr int WAVES_PER_ROW = NCOLS / (16 * TPW);
  const int lane = threadIdx.x & 31;
  const int wave = threadIdx.x >> 5;
  const int rowsPerBlock = (int)(blockDim.x >> 5) / WAVES_PER_ROW;
  const int rowTile = blockIdx.x * rowsPerBlock + wave / WAVES_PER_ROW;
  if (rowTile >= nRowTiles) return;          // wave-uniform: EXEC stays all-1s
  const int tile0 = (wave % WAVES_PER_ROW) * TPW;
  const int m = lane & 15;
  const int half = lane >> 4;

  const float* hrow = H + (size_t)((rowTile << 4) + m) * 128 + (half << 1);
  const float* wp   = Wp + (((size_t)tile0 << 10) + (lane << 1));

  v8f zero = {0.f, 0.f, 0.f, 0.f, 0.f, 0.f, 0.f, 0.f};
  v8f acc[TPW];
#pragma unroll
  for (int t = 0; t < TPW; ++t) acc[t] = zero;

#pragma unroll 8
  for (int ks = 0; ks < 32; ++ks) {
    if ((ks & 7) == 0)                       // stream-ahead on the H row
      __builtin_prefetch(hrow + (ks << 2) + 64, 0, 1);
    v2f a = *(const v2f*)(hrow + (ks << 2));
#pragma unroll
    for (int t = 0; t < TPW; ++t) {
      v2f b = *(const v2f*)(wp + ((size_t)t << 11) + (ks << 6));
      // 8 args: (neg_a, A, neg_b, B, c_mod, C, reuse_a, reuse_b)
      acc[t] = __builtin_amdgcn_wmma_f32_16x16x4_f32(false, a, false, b,
                                                     (short)0, acc[t],
                                                     false, false);
    }
  }

  const int rowBase = (rowTile << 4) + (half << 3);
#pragma unroll
  for (int t = 0; t < TPW; ++t) {
    float* o = Out + (size_t)rowBase * NCOLS + ((tile0 + t) << 4) + m;
#pragma unroll
    for (int v = 0; v < 8; ++v)
      o[(size_t)v * NCOLS] = acc[t][v];
  }
}

// ---------------------------------------------------------------------------
// Utility kernels
// ---------------------------------------------------------------------------
__global__ void fill_f32(float* __restrict__ p, float v, int n) {
  int i = blockIdx.x * blockDim.x + threadIdx.x;
  if (i < n) p[i] = v;
}

// deg[dst[e]] += 1  (deg pre-filled with 1.0 for the self loop; integer-valued
// f32 adds are order-independent -> deterministic)
__global__ void deg_accum(const int* __restrict__ dst, float* __restrict__ deg,
                          int nE) {
  int e = blockIdx.x * blockDim.x + threadIdx.x;
  if (e < nE) atomicAdd(&deg[dst[e]], 1.0f);
}

// in-place: deg -> deg^{-1/2}
__global__ void deg_to_dinv(float* __restrict__ deg, int n) {
  int i = blockIdx.x * blockDim.x + threadIdx.x;
  if (i < n) {
    float d = deg[i];
    deg[i] = (d > 0.f) ? rsqrtf(d) : 0.f;
  }
}

// norm[e] = dinv[src[e]] * dinv[dst[e]]
__global__ void edge_norm(const int* __restrict__ src, const int* __restrict__ dst,
                          const float* __restrict__ dinv, float* __restrict__ norm,
                          int nE) {
  int e = blockIdx.x * blockDim.x + threadIdx.x;
  if (e < nE) norm[e] = dinv[src[e]] * dinv[dst[e]];
}

// agg[dst[e]][c] += hW[src[e]][c] * norm[e]; 4 channels per lane (float4)
// gshift = log2(C/4): 5 for C=128, 2 for C=16.
__global__ void aggregate_edges(const float* __restrict__ hW,
                                const int* __restrict__ src,
                                const int* __restrict__ dst,
                                const float* __restrict__ norm,
                                float* __restrict__ agg,
                                int nE, int gshift, int C) {
  int gid = blockIdx.x * blockDim.x + threadIdx.x;
  int e = gid >> gshift;
  if (e >= nE) return;
  int g = gid & ((1 << gshift) - 1);
  int s = src[e];
  int d = dst[e];
  float nrm = norm[e];
  const float4 v = *(const float4*)(hW + (size_t)s * C + (g << 2));
  float* dp = agg + (size_t)d * C + (g << 2);
  atomicAdd(dp + 0, v.x * nrm);
  atomicAdd(dp + 1, v.y * nrm);
  atomicAdd(dp + 2, v.z * nrm);
  atomicAdd(dp + 3, v.w * nrm);
}

// In-place epilogue: agg += hW * dinv^2 (self-loop) + bias ; optional ReLU.
__global__ void bias_selfloop_act(float* __restrict__ agg,
                                  const float* __restrict__ hW,
                                  const float* __restrict__ dinv,
                                  const float* __restrict__ bias,
                                  int total, int cshift, int doRelu) {
  int i = blockIdx.x * blockDim.x + threadIdx.x;
  if (i >= total) return;
  int node = i >> cshift;
  int c = i & ((1 << cshift) - 1);
  float di = dinv[node];
  float v = agg[i] + hW[i] * (di * di) + bias[c];
  agg[i] = doRelu ? fmaxf(v, 0.f) : v;
}

// ---------------------------------------------------------------------------
// Host-side orchestration (graph-capture safe: launches + memsetAsync only)
// ---------------------------------------------------------------------------
extern "C" void kernel_launch(void* const* d_in, const int* in_sizes, int n_in,
                              void* d_out, int out_size, void* d_ws, size_t ws_size,
                              hipStream_t stream) {
  const float* x     = (const float*)d_in[0];   // [N, 128]
  const int*   ei    = (const int*)d_in[1];     // [2, E]
  const float* Ws    = (const float*)d_in[2];   // [9, 128, 128]
  const float* bs    = (const float*)d_in[3];   // [9, 128]
  const float* W_out = (const float*)d_in[4];   // [128, 16]
  const float* b_out = (const float*)d_in[5];   // [16]
  float* out = (float*)d_out;                   // [N, 16]

  const int D = 128;
  const int N = in_sizes[0] / D;                // 100000
  const int E = in_sizes[1] / 2;                // 600000
  const int C_OUT = in_sizes[5];                // 16

  const int* src = ei;
  const int* dst = ei + E;

  // Workspace: 3 ping-pong feature buffers + dinv + edge norm + packed W
  float* bufs[3];
  bufs[0] = (float*)d_ws;                       // N*D each
  bufs[1] = bufs[0] + (size_t)N * D;
  bufs[2] = bufs[1] + (size_t)N * D;
  float* dinv  = bufs[2] + (size_t)N * D;       // N  (holds deg first)
  float* norm  = dinv + N;                      // E
  float* wpack = norm + E;                      // 128*128 (largest packed W)

  const int TB = 256;

  // ---- degree / normalization (once) ----
  fill_f32<<<(N + TB - 1) / TB, TB, 0, stream>>>(dinv, 1.0f, N);
  deg_accum<<<(E + TB - 1) / TB, TB, 0, stream>>>(dst, dinv, E);
  deg_to_dinv<<<(N + TB - 1) / TB, TB, 0, stream>>>(dinv, N);
  edge_norm<<<(E + TB - 1) / TB, TB, 0, stream>>>(src, dst, dinv, norm, E);

  const int rowTiles = N / 16;                  // 6250
  // <4,128>: 4 waves/block, WAVES_PER_ROW=2 -> 2 row tiles per block
  const int gemmGrid128 = (rowTiles + 1) / 2;
  // <1,16>: 4 waves/block, WAVES_PER_ROW=1 -> 4 row tiles per block
  const int gemmGridOut = (rowTiles + 3) / 4;

  const int aggThreads128 = E * (D / 4);
  const int epiThreads128 = N * D;

  // ---- 9 hidden GCN layers ----
  const float* h = x;
  int hIdx = -1;                                // index of h in bufs (-1 = input x)
  for (int l = 0; l < 9; ++l) {
    int ia = 0; while (ia == hIdx) ++ia;                 // hW buffer
    int ib = 0; while (ib == hIdx || ib == ia) ++ib;     // agg buffer
    float* hW  = bufs[ia];
    float* agg = bufs[ib];

    pack_W_frag<<<(8 * 1024 + TB - 1) / TB, TB, 0, stream>>>(
        Ws + (size_t)l * D * D, wpack, D);
    gcn_gemm_wmma_f32<4, 128><<<gemmGrid128, 128, 0, stream>>>(h, wpack, hW,
                                                               rowTiles);
    hipMemsetAsync(agg, 0, (size_t)N * D * sizeof(float), stream);
    aggregate_edges<<<(aggThreads128 + TB - 1) / TB, TB, 0, stream>>>(
        hW, src, dst, norm, agg, E, 5, D);
    bias_selfloop_act<<<(epiThreads128 + TB - 1) / TB, TB, 0, stream>>>(
        agg, hW, dinv, bs + (size_t)l * D, epiThreads128, 7, 1);

    h = agg;
    hIdx = ib;
  }

  // ---- output layer: GEMM (128x16), aggregate into d_out, bias, no ReLU ----
  int ia = 0; while (ia == hIdx) ++ia;
  float* hW_out = bufs[ia];                     // N x 16
  pack_W_frag<<<(1024 + TB - 1) / TB, TB, 0, stream>>>(W_out, wpack, C_OUT);
  gcn_gemm_wmma_f32<1, 16><<<gemmGridOut, 128, 0, stream>>>(h, wpack, hW_out,
                                                            rowTiles);
  hipMemsetAsync(out, 0, (size_t)N * C_OUT * sizeof(float), stream);
  const int aggThreadsOut = E * (C_OUT / 4);
  aggregate_edges<<<(aggThreadsOut + TB - 1) / TB, TB, 0, stream>>>(
      hW_out, src, dst, norm, out, E, 2, C_OUT);
  const int epiThreadsOut = N * C_OUT;
  bias_selfloop_act<<<(epiThreadsOut + TB - 1) / TB, TB, 0, stream>>>(
      out, hW_out, dinv, b_out, epiThreadsOut, 4, 0);

  (void)n_in; (void)out_size; (void)ws_size;
}